// AffineAdapter_42958262895415
// MI455X (gfx1250) — compile-verified
//
#include <hip/hip_runtime.h>
#include <hip/hip_bf16.h>

typedef float v2f __attribute__((ext_vector_type(2)));
typedef float v8f __attribute__((ext_vector_type(8)));

#define GRIDN 25
#define HW    512
#define HB    511   // blurred spatial size (512 + 2*3 - 8 + 1)
#define NPAD  32    // padded 25 -> 32 (two 16-wide WMMA tiles)

__device__ __forceinline__ float softplus_f(float v) {
    return (v > 20.f) ? v : log1pf(expf(v));
}

// Build the fused operator matrices on device:
//   A  [32][512]  row j: vertical gaussian ⊗ bilinear-y ⊗ validity ⊗ zero-pad
//   BT [512][32]  col i: horizontal equivalent, stored K-major for WMMA B-operand
__global__ void AffineAdapter_build_AB(const float* __restrict__ log_sigma,
                                       const float* __restrict__ log_scale,
                                       float* __restrict__ Amat,
                                       float* __restrict__ BT) {
    const int tid = threadIdx.x;
    for (int i = tid; i < NPAD * HW + HW * NPAD; i += blockDim.x) {
        if (i < NPAD * HW) Amat[i] = 0.f;
        else               BT[i - NPAD * HW] = 0.f;
    }
    __syncthreads();

    const float s0 = softplus_f(log_scale[0]);
    const float s1 = softplus_f(log_scale[1]);
    const float smin = fminf(s0, s1);
    const float sigma_min = (smin >= 1.f) ? 0.f : 0.44f * (1.f / smin - 1.f);
    const float sp = softplus_f(log_sigma[0]);
    const float sigma = sqrtf(sigma_min * sigma_min + sp * sp);

    // torch-quirk taps: arange(-4, 4) -> 8 taps, tap t -> (t-4)
    float g[8]; float gs = 0.f;
    #pragma unroll
    for (int t = 0; t < 8; ++t) {
        float tp = (float)(t - 4);
        g[t] = expf(-0.5f * (tp / sigma) * (tp / sigma));
        gs += g[t];
    }
    #pragma unroll
    for (int t = 0; t < 8; ++t) g[t] /= gs;

    if (tid < GRIDN) {                       // rows of A (y axis, scale[1])
        const int j = tid;
        const float gy = -1.f + 2.f * (float)j / (float)(GRIDN - 1);
        const float iy = (gy * s1 + 1.f) * 0.5f * (float)(HB - 1);
        const float y0f = floorf(iy);
        const float w1 = iy - y0f;
        const float wv[2] = {1.f - w1, w1};
        for (int a = 0; a < 2; ++a) {
            const float yy = y0f + (float)a;
            if (yy >= 0.f && yy <= (float)(HB - 1)) {     // grid_sample 'zeros' mask
                const int y0i = (int)yy;
                for (int t = 0; t < 8; ++t) {
                    const int r = y0i + t - 3;            // conv padding = 3
                    if (r >= 0 && r < HW) Amat[j * HW + r] += wv[a] * g[t];
                }
            }
        }
    } else if (tid >= 32 && tid < 32 + GRIDN) {  // cols of B (x axis, scale[0])
        const int i = tid - 32;
        const float gx = -1.f + 2.f * (float)i / (float)(GRIDN - 1);
        const float ix = (gx * s0 + 1.f) * 0.5f * (float)(HB - 1);
        const float x0f = floorf(ix);
        const float w1 = ix - x0f;
        const float wv[2] = {1.f - w1, w1};
        for (int b = 0; b < 2; ++b) {
            const float xx = x0f + (float)b;
            if (xx >= 0.f && xx <= (float)(HB - 1)) {
                const int x0i = (int)xx;
                for (int t = 0; t < 8; ++t) {
                    const int c = x0i + t - 3;
                    if (c >= 0 && c < HW) BT[c * NPAD + i] += wv[b] * g[t];
                }
            }
        }
    }
}

// One workgroup (8 wave32) per (B,C) image:
//   Stage 1: T(512x32) = x(512x512) * BT   -> LDS (64 KB)
//   Stage 2: out(25x25) = A(32x512) * T
__global__ void __launch_bounds__(256)
AffineAdapter_wmma_fused(const float* __restrict__ x,
                         const float* __restrict__ Amat,
                         const float* __restrict__ BT,
                         float* __restrict__ out) {
    __shared__ float T[HW * NPAD];   // 512*32*4 = 64 KB of the 320 KB WGP LDS

    const int img  = blockIdx.x;
    const float* xi = x + (size_t)img * HW * HW;
    const int lane = threadIdx.x & 31;
    const int wave = threadIdx.x >> 5;
    const int l15  = lane & 15;          // A-op row / B-op col within tile
    const int kq   = (lane >> 4) << 1;   // K sub-offset: 0 or 2

    // ---------------- Stage 1: T = x * BT ----------------
    for (int mt = wave; mt < 32; mt += 8) {       // uniform per wave: EXEC all-ones
        const int m0 = mt * 16;
        v8f acc0 = {};
        v8f acc1 = {};
        const float* xrow = xi + (size_t)(m0 + l15) * HW + kq;
        for (int k = 0; k < HW; k += 4) {
            // A-operand: 16x4 tile of x, lane holds 2 consecutive f32 (8B load)
            v2f a = *reinterpret_cast<const v2f*>(xrow + k);
            // B-operand: 4x16 tiles of BT (K-major), for n0=0 and n0=16
            const float* bp = BT + (k + kq) * NPAD + l15;
            v2f b0 = { bp[0],  bp[NPAD]      };
            v2f b1 = { bp[16], bp[NPAD + 16] };
            acc0 = __builtin_amdgcn_wmma_f32_16x16x4_f32(false, a, false, b0,
                                                         (short)0, acc0, false, false);
            acc1 = __builtin_amdgcn_wmma_f32_16x16x4_f32(false, a, false, b1,
                                                         (short)0, acc1, false, false);
        }
        // D layout: vgpr r, lane L -> (m0 + r + 8*(L>>4), n0 + (L&15))
        #pragma unroll
        for (int r = 0; r < 8; ++r) {
            const int row = m0 + r + ((lane >> 4) << 3);
            T[row * NPAD + l15]      = acc0[r];
            T[row * NPAD + 16 + l15] = acc1[r];
        }
    }
    __syncthreads();

    // ---------------- Stage 2: out = A * T ----------------
    if (wave < 4) {                               // whole-wave uniform branch
        const int m0 = (wave >> 1) * 16;
        const int n0 = (wave & 1) * 16;
        v8f acc = {};
        const float* arow = Amat + (size_t)(m0 + l15) * HW + kq;
        for (int k = 0; k < HW; k += 4) {
            v2f a = *reinterpret_cast<const v2f*>(arow + k);
            const float* tp = T + (k + kq) * NPAD + n0 + l15;
            v2f b = { tp[0], tp[NPAD] };
            acc = __builtin_amdgcn_wmma_f32_16x16x4_f32(false, a, false, b,
                                                        (short)0, acc, false, false);
        }
        float* op = out + (size_t)img * GRIDN * GRIDN;
        #pragma unroll
        for (int r = 0; r < 8; ++r) {
            const int j = m0 + r + ((lane >> 4) << 3);
            const int i = n0 + l15;
            if (j < GRIDN && i < GRIDN) op[j * GRIDN + i] = acc[r];
        }
    }
}

extern "C" void kernel_launch(void* const* d_in, const int* in_sizes, int n_in,
                              void* d_out, int out_size, void* d_ws, size_t ws_size,
                              hipStream_t stream) {
    const float* x         = (const float*)d_in[0];   // (16,8,512,512) f32
    const float* log_sigma = (const float*)d_in[1];   // scalar
    const float* log_scale = (const float*)d_in[2];   // (2,)

    float* Amat = (float*)d_ws;          // 32*512 f32 = 64 KB
    float* BT   = Amat + NPAD * HW;      // 512*32 f32 = 64 KB

    AffineAdapter_build_AB<<<1, 64, 0, stream>>>(log_sigma, log_scale, Amat, BT);
    AffineAdapter_wmma_fused<<<16 * 8, 256, 0, stream>>>(x, Amat, BT, (float*)d_out);
}